// RecursiveRegression_35631048688010
// MI455X (gfx1250) — compile-verified
//
#include <hip/hip_runtime.h>

typedef __attribute__((ext_vector_type(2))) float v2f;
typedef __attribute__((ext_vector_type(8))) float v8f;

#define T_LEN 2048
#define START 8      // start_ix = max(H_WINDOW-1, X_WINDOW) = 8
#define NBLK  128    // ceil((T_LEN - START) / 16)

struct LaneCtx {
    v2f   La[4];   // A-operand chunks of L (16x16), lane holds L[col][4k+2hf+t]
    v2f   Ga[2];   // A-operand chunks of G (16x8)
    float hw[8];   // h_weights
    float bias;
    int   hf;      // lane half (0/1)
};

// Shifted window loaded directly: win[q] = h[row][i0-8+2*hf+q], q=0..21.
// Base is always even -> 8B aligned; in-bounds for blocks 0..NBLK-2.
__device__ __forceinline__ void load_win_sh(float win[22], const float* __restrict__ hrow,
                                            int i0, int hf)
{
    const float* __restrict__ p = hrow + (i0 - 8) + 2 * hf;
    #pragma unroll
    for (int q = 0; q < 11; ++q) {
        float2 r = *(const float2*)(p + 2 * q);
        win[2 * q + 0] = r.x;
        win[2 * q + 1] = r.y;
    }
}

// Tail version: per-element clamped scalar loads (last block reads past T-1).
__device__ __forceinline__ void load_win_clamped(float win[22], const float* __restrict__ hrow,
                                                 int i0, int hf)
{
    #pragma unroll
    for (int q = 0; q < 22; ++q) {
        int idx = i0 - 8 + 2 * hf + q;
        idx = idx < T_LEN ? idx : (T_LEN - 1);
        win[q] = hrow[idx];
    }
}

// One 16-step block: Out = L*D + G*S via 6x V_WMMA_F32_16X16X4_F32 (two chains).
template <bool TRANSIENT>
__device__ __forceinline__ v8f block_eval(const LaneCtx& cx, const float win[22],
                                          const float s4[4])
{
    // drive terms d_p, p = 4c + 2*hf + t  (bias + 8-tap FIR over h)
    float dv[4][2];
    #pragma unroll
    for (int c = 0; c < 4; ++c) {
        #pragma unroll
        for (int t = 0; t < 2; ++t) {
            float a = cx.bias;
            if (TRANSIENT && c < 2) {
                // steps i=8..15: h-window skips h[8] and h[9]
                const int bound = 8 - 2 * cx.hf - 4 * c - t;   // k < bound -> old part
                #pragma unroll
                for (int k = 0; k < 8; ++k) {
                    float v = (k < bound) ? win[4 * c + t + k] : win[4 * c + t + k + 2];
                    a = fmaf(cx.hw[k], v, a);
                }
            } else {
                #pragma unroll
                for (int k = 0; k < 8; ++k)
                    a = fmaf(cx.hw[k], win[4 * c + t + 1 + k], a);
            }
            dv[c][t] = a;
        }
    }

    v8f accL = {0.f, 0.f, 0.f, 0.f, 0.f, 0.f, 0.f, 0.f};
    #pragma unroll
    for (int k = 0; k < 4; ++k) {
        v2f bb; bb.x = dv[k][0]; bb.y = dv[k][1];
        accL = __builtin_amdgcn_wmma_f32_16x16x4_f32(
            false, cx.La[k], false, bb, (short)0, accL, false, false);
    }
    v8f accG = {0.f, 0.f, 0.f, 0.f, 0.f, 0.f, 0.f, 0.f};
    #pragma unroll
    for (int k = 0; k < 2; ++k) {
        v2f bb; bb.x = s4[2 * k]; bb.y = s4[2 * k + 1];  // S rows 4k+t+2hf
        accG = __builtin_amdgcn_wmma_f32_16x16x4_f32(
            false, cx.Ga[k], false, bb, (short)0, accG, false, false);
    }
    #pragma unroll
    for (int i = 0; i < 8; ++i) accL[i] += accG[i];
    return accL;
}

// New state = Out rows 8..15. Half-1 lanes already own them (rows r+8 live in
// lanes 16..31); half-0 lanes fetch the 4 entries they need via shuffle.
__device__ __forceinline__ void update_state(float s4[4], const v8f& a, int hf, int srcLane)
{
    float t0 = __shfl(a[0], srcLane, 32);
    float t1 = __shfl(a[1], srcLane, 32);
    float t2 = __shfl(a[4], srcLane, 32);
    float t3 = __shfl(a[5], srcLane, 32);
    s4[0] = hf ? a[2] : t0;
    s4[1] = hf ? a[3] : t1;
    s4[2] = hf ? a[6] : t2;
    s4[3] = hf ? a[7] : t3;
}

__device__ __forceinline__ void store_out(float* __restrict__ orow, int i0, int hf, const v8f& a)
{
    const int tbase = i0 + 8 * hf;                 // 32B aligned
    float4 o0 = { a[0], a[1], a[2], a[3] };
    float4 o1 = { a[4], a[5], a[6], a[7] };
    *(float4*)(orow + tbase)     = o0;
    *(float4*)(orow + tbase + 4) = o1;
}

__global__ __launch_bounds__(256) void rr_blockscan_wmma(
    const float* __restrict__ hbuf,
    const float* __restrict__ h_weights,
    const float* __restrict__ x_weights,
    const float* __restrict__ biases,
    float* __restrict__ out)
{
    __shared__ float Lmat[16][16];  // in-block impulse response (lower tri, unit diag)
    __shared__ float Gmat[16][8];   // state propagation

    const int tid  = threadIdx.x;
    const int lane = tid & 31;
    const int wave = tid >> 5;
    const int hf   = lane >> 4;
    const int col  = lane & 15;

    // ---- build L and G once per workgroup (threads 0..15 -> L cols, 16..23 -> G cols)
    if (tid < 24) {
        float c[8];
        #pragma unroll
        for (int m = 0; m < 8; ++m) c[m] = x_weights[m];
        float vals[24];                           // vals[0..7] = incoming state
        #pragma unroll
        for (int m = 0; m < 8; ++m)
            vals[m] = (tid >= 16 && (tid - 16) == m) ? 1.0f : 0.0f;
        #pragma unroll
        for (int j = 0; j < 16; ++j) {
            float a = (tid < 16 && j == tid) ? 1.0f : 0.0f;  // unit drive impulse
            #pragma unroll
            for (int m = 0; m < 8; ++m) a = fmaf(c[m], vals[j + m], a);
            vals[8 + j] = a;
            if (tid < 16) Lmat[j][tid] = a;
            else          Gmat[j][tid - 16] = a;
        }
    }
    __syncthreads();

    LaneCtx cx;
    cx.hf = hf;
    #pragma unroll
    for (int k = 0; k < 8; ++k) cx.hw[k] = h_weights[k];
    cx.bias = biases[0];
    #pragma unroll
    for (int k = 0; k < 4; ++k) {
        cx.La[k].x = Lmat[col][4 * k + 2 * hf + 0];
        cx.La[k].y = Lmat[col][4 * k + 2 * hf + 1];
    }
    #pragma unroll
    for (int k = 0; k < 2; ++k) {
        cx.Ga[k].x = Gmat[col][4 * k + 2 * hf + 0];
        cx.Ga[k].y = Gmat[col][4 * k + 2 * hf + 1];
    }

    const int tile = blockIdx.x * 8 + wave;        // 512 tiles, one per wave
    const int row  = tile * 16 + col;
    const float* __restrict__ hrow = hbuf + (size_t)row * T_LEN;
    float* __restrict__       orow = out  + (size_t)row * T_LEN;

    // out[:, 0:8] = 0
    if (hf == 0) {
        float4 z = {0.f, 0.f, 0.f, 0.f};
        *(float4*)(orow + 0) = z;
        *(float4*)(orow + 4) = z;
    }

    const int srcLane = 16 + col;
    float s4[4] = {0.f, 0.f, 0.f, 0.f};           // x-window starts as zeros

    // ---- block 0: transient h-window ----
    {
        float win[22];
        load_win_sh(win, hrow, START, hf);
        v8f a = block_eval<true>(cx, win, s4);
        store_out(orow, START, hf, a);
        update_state(s4, a, hf, srcLane);
    }

    // ---- steady blocks 1..126 ----
    for (int blk = 1; blk < NBLK - 1; ++blk) {
        const int i0 = START + 16 * blk;
        float win[22];
        load_win_sh(win, hrow, i0, hf);
        v8f a = block_eval<false>(cx, win, s4);
        store_out(orow, i0, hf, a);
        update_state(s4, a, hf, srcLane);
    }

    // ---- final block (i0 = 2040): 8 valid steps, clamped loads, half-0 stores ----
    {
        const int i0 = START + 16 * (NBLK - 1);
        float win[22];
        load_win_clamped(win, hrow, i0, hf);
        v8f a = block_eval<false>(cx, win, s4);
        if (hf == 0) store_out(orow, i0, 0, a);
    }
}

extern "C" void kernel_launch(void* const* d_in, const int* in_sizes, int n_in,
                              void* d_out, int out_size, void* d_ws, size_t ws_size,
                              hipStream_t stream) {
    const float* h   = (const float*)d_in[0];
    const float* hwv = (const float*)d_in[1];
    const float* xwv = (const float*)d_in[2];
    const float* bi  = (const float*)d_in[3];
    float* out = (float*)d_out;

    const int B      = in_sizes[0] / T_LEN;  // 8192
    const int tiles  = B / 16;               // one 16-row tile per wave
    const int blocks = tiles / 8;            // 8 waves per workgroup

    rr_blockscan_wmma<<<blocks, 256, 0, stream>>>(h, hwv, xwv, bi, out);
}